// HyperbolicGraphConvolution_541165879720
// MI455X (gfx1250) — compile-verified
//
#include <hip/hip_runtime.h>

typedef float v2f __attribute__((ext_vector_type(2)));
typedef float v8f __attribute__((ext_vector_type(8)));

#define MIN_NORM 1e-15f
#define MAXNORM  0.996f   // (1 - 4e-3)/sqrt(c), c = 1
#define D 128

__device__ __forceinline__ float waveReduceSum(float v) {
#pragma unroll
    for (int off = 16; off > 0; off >>= 1)
        v += __shfl_xor(v, off, 32);
    return v;
}

__device__ __forceinline__ float artanh_clip(float x) {
    x = fminf(fmaxf(x, -1.0f + 1e-7f), 1.0f - 1e-7f);
    return 0.5f * (log1pf(x) - log1pf(-x));
}

// ---------------------------------------------------------------------------
// Kernel 0: zero the aggregation buffer (d_out doubles as segment-sum acc)
// ---------------------------------------------------------------------------
__global__ __launch_bounds__(256) void zero_kernel(float* __restrict__ p, int n4) {
    int i = blockIdx.x * blockDim.x + threadIdx.x;
    if (i < n4) {
        float4 z = {0.f, 0.f, 0.f, 0.f};
        *(float4*)(p + (size_t)i * 4) = z;
    }
}

// ---------------------------------------------------------------------------
// Kernel 1: fused HypLinear (mobius_matvec + proj + mobius_add(bias) + proj)
//           followed by logmap0  ->  xt  [N, 128]
// Block = 128 threads = 4 waves; each wave owns a 16-row tile.
// GEMM: mx = x @ W^T via V_WMMA_F32_16X16X4_F32, K = 128 in steps of 4.
// OOB rows are handled by clamping the A-row index (results discarded in
// phase 2), so the hot loop has no exec-mask divergence.
// ---------------------------------------------------------------------------
__global__ __launch_bounds__(128)
void hyp_linear_logmap_kernel(const float* __restrict__ x,
                              const float* __restrict__ W,
                              const float* __restrict__ b,
                              float* __restrict__ xt,
                              int N) {
    __shared__ float ldsTile[64][132];   // 64 rows x 128 cols (+pad)
    __shared__ float ldsBias[D];
    __shared__ float ldsBias2;           // ||hyp_bias||^2

    const int tid  = threadIdx.x;
    const int wave = tid >> 5;
    const int lane = tid & 31;

    // --- wave 0: hyp_bias = proj(expmap0(b)) ------------------------------
    if (wave == 0) {
        float4 bv = *(const float4*)(b + lane * 4);
        float nb2 = waveReduceSum(bv.x * bv.x + bv.y * bv.y + bv.z * bv.z + bv.w * bv.w);
        float nb  = fmaxf(sqrtf(nb2), MIN_NORM);
        float en  = tanhf(nb);           // norm of expmap0(b)
        float s   = en / nb;             // expmap0 scale
        if (en > MAXNORM) s *= MAXNORM / en;
        float fn = fminf(en, MAXNORM);
        bv.x *= s; bv.y *= s; bv.z *= s; bv.w *= s;
        *(float4*)(&ldsBias[lane * 4]) = bv;
        if (lane == 0) ldsBias2 = fn * fn;
    }

    // --- GEMM: 16 rows x 128 cols per wave --------------------------------
    const int rowBase = blockIdx.x * 64 + wave * 16;
    const int m    = lane & 15;          // M (and N) index within tile
    const int koff = (lane >> 4) * 2;    // K sub-offset per A/B layout
    const int arow = rowBase + m;
    const int arowc = (arow < N) ? arow : (N - 1);   // clamp: padding rows unused

    const float* __restrict__ xr = x + (size_t)arowc * D + koff;
    const float* __restrict__ wr = W + (size_t)m * D + koff;

    v8f acc[8] = {};
    for (int kk = 0; kk < D; kk += 4) {
        const float2 xa = *(const float2*)(xr + kk);
        v2f a; a[0] = xa.x; a[1] = xa.y;
#pragma unroll
        for (int j = 0; j < 8; ++j) {
            // B[k][n] = W[n][k]  (mx = x @ W^T)
            const float2 wb = *(const float2*)(wr + (size_t)j * 16 * D + kk);
            v2f bm; bm[0] = wb.x; bm[1] = wb.y;
            acc[j] = __builtin_amdgcn_wmma_f32_16x16x4_f32(
                false, a, false, bm, (short)0, acc[j], false, false);
        }
    }

    // --- spill C tiles to LDS (lane-layout -> row-major) ------------------
    {
        const int mhi = (lane >> 4) * 8; // VGPR r holds row r (lanes 0-15) / r+8 (16-31)
#pragma unroll
        for (int j = 0; j < 8; ++j) {
#pragma unroll
            for (int r = 0; r < 8; ++r) {
                ldsTile[wave * 16 + mhi + r][j * 16 + m] = acc[j][r];
            }
        }
    }
    __syncthreads();

    // --- per-row mobius pipeline: one wave per row, 4 floats per lane -----
    for (int lr = 0; lr < 16; ++lr) {
        const int grow = rowBase + lr;
        if (grow >= N) continue;   // uniform across wave

        float4 mx = *(float4*)(&ldsTile[wave * 16 + lr][lane * 4]);
        float4 xv = *(const float4*)(x + (size_t)grow * D + lane * 4);

        float xn2  = waveReduceSum(xv.x * xv.x + xv.y * xv.y + xv.z * xv.z + xv.w * xv.w);
        float mxn2 = waveReduceSum(mx.x * mx.x + mx.y * mx.y + mx.z * mx.z + mx.w * mx.w);
        float xn     = fmaxf(sqrtf(xn2), MIN_NORM);
        float mxnraw = sqrtf(mxn2);
        float mxn    = fmaxf(mxnraw, MIN_NORM);

        // mobius_matvec tail:  res = tanh(mxn/xn * artanh(xn)) * mx / mxn
        float alpha = tanhf(mxn / xn * artanh_clip(xn)) / mxn;
        if (mxnraw == 0.0f) alpha = 0.0f;
        float4 h = {mx.x * alpha, mx.y * alpha, mx.z * alpha, mx.w * alpha};

        // proj
        float hn = fmaxf(alpha * mxnraw, MIN_NORM);
        if (hn > MAXNORM) {
            float sc = MAXNORM / hn;
            h.x *= sc; h.y *= sc; h.z *= sc; h.w *= sc;
            hn = MAXNORM;
        }

        // mobius_add(h, hyp_bias)
        float4 bias = *(float4*)(&ldsBias[lane * 4]);
        float xy = waveReduceSum(h.x * bias.x + h.y * bias.y + h.z * bias.z + h.w * bias.w);
        float x2 = hn * hn;
        float y2 = ldsBias2;
        float ca  = 1.0f + 2.0f * xy + y2;
        float cb  = 1.0f - x2;
        float den = fmaxf(1.0f + 2.0f * xy + x2 * y2, MIN_NORM);
        float inv = 1.0f / den;
        float4 sv = {(ca * h.x + cb * bias.x) * inv,
                     (ca * h.y + cb * bias.y) * inv,
                     (ca * h.z + cb * bias.z) * inv,
                     (ca * h.w + cb * bias.w) * inv};

        // proj
        float sn2 = waveReduceSum(sv.x * sv.x + sv.y * sv.y + sv.z * sv.z + sv.w * sv.w);
        float sn  = fmaxf(sqrtf(sn2), MIN_NORM);
        if (sn > MAXNORM) {
            float sc = MAXNORM / sn;
            sv.x *= sc; sv.y *= sc; sv.z *= sc; sv.w *= sc;
            sn = MAXNORM;
        }

        // logmap0
        float ls = artanh_clip(sn) / sn;
        float4 o = {sv.x * ls, sv.y * ls, sv.z * ls, sv.w * ls};
        *(float4*)(xt + (size_t)grow * D + lane * 4) = o;
    }
}

// ---------------------------------------------------------------------------
// Kernel 2: edge scatter  agg[dst] += edge_w * xt[src]
// One wave per edge; 32 lanes x float4 = 128 columns.
// ---------------------------------------------------------------------------
__global__ __launch_bounds__(256)
void edge_scatter_kernel(const float* __restrict__ xt,
                         const float* __restrict__ edge_w,
                         const int* __restrict__ src,
                         const int* __restrict__ dst,
                         float* __restrict__ agg,
                         int E) {
    const int wid  = (blockIdx.x * blockDim.x + threadIdx.x) >> 5;
    const int lane = threadIdx.x & 31;
    if (wid >= E) return;
    const int   s = src[wid];
    const int   d = dst[wid];
    const float w = edge_w[wid];
    float4 v = *(const float4*)(xt + (size_t)s * D + lane * 4);
    float* o = agg + (size_t)d * D + lane * 4;
    atomicAdd(o + 0, w * v.x);
    atomicAdd(o + 1, w * v.y);
    atomicAdd(o + 2, w * v.z);
    atomicAdd(o + 3, w * v.w);
}

// ---------------------------------------------------------------------------
// Kernel 3: HypAct in-place:  out = proj(expmap0(relu(logmap0(proj(expmap0(agg))))))
// One wave per row.
// ---------------------------------------------------------------------------
__global__ __launch_bounds__(256)
void hyp_act_kernel(float* __restrict__ out, int N) {
    const int wid  = (blockIdx.x * blockDim.x + threadIdx.x) >> 5;
    const int lane = threadIdx.x & 31;
    if (wid >= N) return;
    float* row = out + (size_t)wid * D;
    float4 g = *(float4*)(row + lane * 4);

    // expmap0 + proj
    float n1r = sqrtf(waveReduceSum(g.x * g.x + g.y * g.y + g.z * g.z + g.w * g.w));
    float n1  = fmaxf(n1r, MIN_NORM);
    float e1  = tanhf(n1);
    float s1  = e1 / n1;
    if (e1 > MAXNORM) s1 *= MAXNORM / e1;
    float hn = fminf(e1, MAXNORM);
    float4 h = {g.x * s1, g.y * s1, g.z * s1, g.w * s1};

    // logmap0 + relu
    float n2 = fmaxf(hn, MIN_NORM);
    float l2 = artanh_clip(n2) / n2;
    float4 t = {fmaxf(h.x * l2, 0.0f), fmaxf(h.y * l2, 0.0f),
                fmaxf(h.z * l2, 0.0f), fmaxf(h.w * l2, 0.0f)};

    // expmap0 + proj (c_out = 1)
    float n3 = fmaxf(sqrtf(waveReduceSum(t.x * t.x + t.y * t.y + t.z * t.z + t.w * t.w)),
                     MIN_NORM);
    float e3 = tanhf(n3);
    float s3 = e3 / n3;
    if (e3 > MAXNORM) s3 *= MAXNORM / e3;
    float4 o = {t.x * s3, t.y * s3, t.z * s3, t.w * s3};
    *(float4*)(row + lane * 4) = o;
}

// ---------------------------------------------------------------------------
extern "C" void kernel_launch(void* const* d_in, const int* in_sizes, int n_in,
                              void* d_out, int out_size, void* d_ws, size_t ws_size,
                              hipStream_t stream) {
    const float* x   = (const float*)d_in[0];
    const float* W   = (const float*)d_in[1];
    const float* b   = (const float*)d_in[2];
    const float* ew  = (const float*)d_in[3];
    const int*   src = (const int*)d_in[4];
    const int*   dst = (const int*)d_in[5];

    const int N = in_sizes[0] / D;
    const int E = in_sizes[3];

    float* out = (float*)d_out;   // doubles as segment-sum accumulator
    float* xt  = (float*)d_ws;    // [N, 128] tangent-space features

    // 1) zero accumulator
    const int n4 = (N * D) / 4;
    zero_kernel<<<(n4 + 255) / 256, 256, 0, stream>>>(out, n4);

    // 2) fused HypLinear + logmap0 (WMMA GEMM inside)
    hyp_linear_logmap_kernel<<<(N + 63) / 64, 128, 0, stream>>>(x, W, b, xt, N);

    // 3) edge-weighted scatter-add (one wave per edge)
    const long long threadsE = (long long)E * 32;
    edge_scatter_kernel<<<(int)((threadsE + 255) / 256), 256, 0, stream>>>(
        xt, ew, src, dst, out, E);

    // 4) HypAct in place on d_out (one wave per row)
    const long long threadsN = (long long)N * 32;
    hyp_act_kernel<<<(int)((threadsN + 255) / 256), 256, 0, stream>>>(out, N);
}